// Actor_70282844831800
// MI455X (gfx1250) — compile-verified
//
#include <hip/hip_runtime.h>

typedef float v2f __attribute__((ext_vector_type(2)));
typedef float v8f __attribute__((ext_vector_type(8)));

#define HID   128
#define INDIM 18

__device__ __forceinline__ float leaky02(float x) { return x > 0.f ? x : 0.2f * x; }
__device__ __forceinline__ float eluf(float x)    { return x > 0.f ? x : (__expf(x) - 1.f); }
__device__ __forceinline__ float neg_inf()        { return __int_as_float(0xff800000); }

// Symmetrized edge list + self loops, evaluated implicitly:
//   e <  E      : (src=ei[e],   dst=ei[E+e])
//   E <= e < 2E : (src=ei[e],   dst=ei[e-E])      (flipped copy)
//   e >= 2E     : (src=dst=e-2E)                  (self loop)
__device__ __forceinline__ void edge_sd(int e, const int* __restrict__ ei, int E, int& s, int& d) {
    if (e < E)            { s = ei[e]; d = ei[E + e]; }
    else if (e < 2 * E)   { s = ei[e]; d = ei[e - E]; }
    else                  { s = e - 2 * E; d = s; }
}

// Monotone float atomic max via int/uint punning (amax initialized to -inf).
__device__ __forceinline__ void atomicMaxFloat(float* addr, float v) {
    if (v >= 0.f) atomicMax((int*)addr, __float_as_int(v));
    else          atomicMin((unsigned int*)addr, (unsigned int)__float_as_int(v));
}

__global__ void k_init_node(float* __restrict__ amax, float* __restrict__ denom, int N) {
    int i = blockIdx.x * blockDim.x + threadIdx.x;
    if (i < N) { amax[i] = neg_inf(); denom[i] = 0.f; }
}

__global__ void k_zero(float* __restrict__ p, long n) {
    long i = (long)blockIdx.x * blockDim.x + threadIdx.x;
    long stride = (long)gridDim.x * blockDim.x;
    for (; i < n; i += stride) p[i] = 0.f;
}

// layer0 GEMM: z[N,128] = x[N,18] @ W[18,128]   (K too skinny for WMMA)
__global__ void k_gemm_in(const float* __restrict__ x, const float* __restrict__ W,
                          float* __restrict__ z, int N) {
    __shared__ float sx[INDIM];
    int n = blockIdx.x;
    int c = threadIdx.x;
    if (c < INDIM) sx[c] = x[n * INDIM + c];
    __syncthreads();
    float acc = 0.f;
#pragma unroll
    for (int k = 0; k < INDIM; ++k) acc += sx[k] * W[k * HID + c];
    z[(long)n * HID + c] = acc;
}

// per-node attention dots: es[n] = z[n,:]·a_src ; ed[n] = z[n,:]·a_dst
__global__ void k_dots(const float* __restrict__ z, const float* __restrict__ as,
                       const float* __restrict__ ad, float* __restrict__ es,
                       float* __restrict__ ed, int N) {
    __shared__ float s1[HID];
    __shared__ float s2[HID];
    int n = blockIdx.x, c = threadIdx.x;
    float v = z[(long)n * HID + c];
    s1[c] = v * as[c];
    s2[c] = v * ad[c];
    __syncthreads();
    for (int off = 64; off > 0; off >>= 1) {
        if (c < off) { s1[c] += s1[c + off]; s2[c] += s2[c + off]; }
        __syncthreads();
    }
    if (c == 0) { es[n] = s1[0]; ed[n] = s2[0]; }
}

// edge pass A: segment max of leaky_relu(es[s]+ed[d]) over incoming edges
__global__ void k_edge_max(const int* __restrict__ ei, int E, int N,
                           const float* __restrict__ es, const float* __restrict__ ed,
                           float* __restrict__ amax) {
    int e = blockIdx.x * blockDim.x + threadIdx.x;
    int ET = 2 * E + N;
    if (e >= ET) return;
    int s, d;
    edge_sd(e, ei, E, s, d);
    atomicMaxFloat(&amax[d], leaky02(es[s] + ed[d]));
}

// edge pass B: one wave per edge; denom += ex, accum[d,:] += ex * z[s,:]
__global__ void k_edge_accum(const int* __restrict__ ei, int E, int N,
                             const float* __restrict__ es, const float* __restrict__ ed,
                             const float* __restrict__ amax, const float* __restrict__ z,
                             float* __restrict__ denom, float* __restrict__ accum) {
    int wave = blockIdx.x * (blockDim.x >> 5) + (threadIdx.x >> 5);
    int lane = threadIdx.x & 31;
    int ET = 2 * E + N;
    if (wave >= ET) return;
    int s, d;
    edge_sd(wave, ei, E, s, d);
    float a  = leaky02(es[s] + ed[d]);
    float ex = __expf(a - amax[d]);
    if (lane == 0) atomicAdd(&denom[d], ex);
    const float* zs = z + (long)s * HID;
    float* ac = accum + (long)d * HID;
#pragma unroll
    for (int j = 0; j < 4; ++j) {
        int c = lane + 32 * j;
        atomicAdd(&ac[c], ex * zs[c]);
    }
}

// h[n,c] = elu(accum[n,c]/denom[n] + b[c])
__global__ void k_finalize(const float* __restrict__ accum, const float* __restrict__ denom,
                           const float* __restrict__ b, float* __restrict__ h, int N) {
    long i = (long)blockIdx.x * blockDim.x + threadIdx.x;
    long total = (long)N * HID;
    if (i >= total) return;
    int n = (int)(i >> 7);
    int c = (int)(i & 127);
    h[i] = eluf(accum[i] / denom[n] + b[c]);
}

// layer1 GEMM: z[N,128] = h[N,128] @ W[128,128] with V_WMMA_F32_16X16X4_F32.
// One wave per 16x16 output tile; 32 K-steps of 4.
// A (16x4 f32): lanes 0-15 -> M=lane, {K0,K1}; lanes 16-31 -> M=lane-16, {K2,K3}.
// B (4x16 f32): lanes 0-15 -> N=lane, {K0,K1}; lanes 16-31 -> N=lane-16, {K2,K3}.
// D (16x16 f32): VGPR r holds rows r (lanes 0-15) and r+8 (lanes 16-31).
__global__ void k_gemm128_wmma(const float* __restrict__ h, const float* __restrict__ W,
                               float* __restrict__ z, int N) {
    int tileN = threadIdx.x >> 5;   // wave id 0..7 -> 16-col slab
    int lane  = threadIdx.x & 31;
    int tileM = blockIdx.x;
    int half  = lane >> 4;
    int idx   = lane & 15;
    int row   = tileM * 16 + idx;
    int rcl   = row < N ? row : 0;  // clamp loads; stores guarded below
    int col   = tileN * 16 + idx;
    const float* hrow = h + (long)rcl * HID;
    v8f c = {};
#pragma unroll
    for (int s = 0; s < 32; ++s) {
        int k = 4 * s + 2 * half;
        v2f a, bf;
        a.x  = hrow[k];
        a.y  = hrow[k + 1];
        bf.x = W[k * HID + col];
        bf.y = W[(k + 1) * HID + col];
        c = __builtin_amdgcn_wmma_f32_16x16x4_f32(false, a, false, bf, (short)0, c, false, false);
    }
#pragma unroll
    for (int r = 0; r < 8; ++r) {
        int orow = tileM * 16 + r + 8 * half;
        if (orow < N) z[(long)orow * HID + col] = c[r];
    }
}

// Fused actor MLP: scores[m] = relu(h[cidx[m],:] @ mW0 + mb0) @ mW1 + mb1
// 16 candidates per block; wave j computes hidden cols 16j..16j+15 via WMMA,
// then ReLU + dot with mW1 slice, reduced through LDS float atomics.
__global__ void k_mlp_wmma(const float* __restrict__ h, const int* __restrict__ cidx,
                           const float* __restrict__ mW0, const float* __restrict__ mb0,
                           const float* __restrict__ mW1, const float* __restrict__ mb1,
                           float* __restrict__ scores, int BV) {
    __shared__ float ssc[16];
    int wid  = threadIdx.x >> 5;
    int lane = threadIdx.x & 31;
    int half = lane >> 4;
    int idx  = lane & 15;
    int base = blockIdx.x * 16;
    if (threadIdx.x < 16) ssc[threadIdx.x] = 0.f;
    __syncthreads();

    int m    = base + idx;
    int node = (m < BV) ? cidx[m] : 0;
    const float* hrow = h + (long)node * HID;
    int col = wid * 16 + idx;
    v8f c = {};
#pragma unroll
    for (int s = 0; s < 32; ++s) {
        int k = 4 * s + 2 * half;
        v2f a, bf;
        a.x  = hrow[k];
        a.y  = hrow[k + 1];
        bf.x = mW0[k * HID + col];
        bf.y = mW0[(k + 1) * HID + col];
        c = __builtin_amdgcn_wmma_f32_16x16x4_f32(false, a, false, bf, (short)0, c, false, false);
    }
    float w1  = mW1[col];
    float b0c = mb0[col];
#pragma unroll
    for (int r = 0; r < 8; ++r) {
        int   mrow = r + 8 * half;
        float hv   = c[r] + b0c;
        hv = hv > 0.f ? hv : 0.f;
        atomicAdd(&ssc[mrow], hv * w1);
    }
    __syncthreads();
    if (threadIdx.x < 16 && base + threadIdx.x < BV)
        scores[base + threadIdx.x] = ssc[threadIdx.x] + mb1[0];
}

// softmax over groups of 16 consecutive scores
__global__ void k_softmax16(const float* __restrict__ scores, float* __restrict__ out, int G) {
    int g = blockIdx.x * blockDim.x + threadIdx.x;
    if (g >= G) return;
    float v[16];
    float mx = neg_inf();
#pragma unroll
    for (int j = 0; j < 16; ++j) { v[j] = scores[g * 16 + j]; mx = fmaxf(mx, v[j]); }
    float sum = 0.f;
#pragma unroll
    for (int j = 0; j < 16; ++j) { v[j] = __expf(v[j] - mx); sum += v[j]; }
    float inv = 1.f / sum;
#pragma unroll
    for (int j = 0; j < 16; ++j) out[g * 16 + j] = v[j] * inv;
}

extern "C" void kernel_launch(void* const* d_in, const int* in_sizes, int n_in,
                              void* d_out, int out_size, void* d_ws, size_t ws_size,
                              hipStream_t stream) {
    const float* state_wf = (const float*)d_in[0];
    const int*   ei       = (const int*)d_in[1];
    const int*   cidx     = (const int*)d_in[2];
    const float* W0  = (const float*)d_in[3];
    const float* as0 = (const float*)d_in[4];
    const float* ad0 = (const float*)d_in[5];
    const float* b0  = (const float*)d_in[6];
    const float* W1  = (const float*)d_in[7];
    const float* as1 = (const float*)d_in[8];
    const float* ad1 = (const float*)d_in[9];
    const float* b1  = (const float*)d_in[10];
    const float* mW0 = (const float*)d_in[11];
    const float* mb0 = (const float*)d_in[12];
    const float* mW1 = (const float*)d_in[13];
    const float* mb1 = (const float*)d_in[14];

    int N  = in_sizes[0] / INDIM;   // 100000
    int E  = in_sizes[1] / 2;       // 800000
    int BV = in_sizes[2];           // 65536
    int ET = 2 * E + N;

    long   nh   = (long)N * HID;
    float* ws   = (float*)d_ws;
    float* bufZ = ws;               // z / accum / h ping-pong
    float* bufA = ws + nh;
    float* es    = ws + 2 * nh;
    float* ed    = es + N;
    float* amax  = ed + N;
    float* denom = amax + N;
    float* scores = denom + N;      // BV floats

    int eb = (ET + 255) / 256;
    int wb = (ET + 7) / 8;          // wave-per-edge blocks (8 waves/block)
    int fb = (int)((nh + 255) / 256);

    // ---------------- GAT layer 0 ----------------
    k_init_node<<<(N + 255) / 256, 256, 0, stream>>>(amax, denom, N);
    k_zero<<<2048, 256, 0, stream>>>(bufA, nh);                              // accum0
    k_gemm_in<<<N, HID, 0, stream>>>(state_wf, W0, bufZ, N);                 // z0
    k_dots<<<N, HID, 0, stream>>>(bufZ, as0, ad0, es, ed, N);
    k_edge_max<<<eb, 256, 0, stream>>>(ei, E, N, es, ed, amax);
    k_edge_accum<<<wb, 256, 0, stream>>>(ei, E, N, es, ed, amax, bufZ, denom, bufA);
    k_finalize<<<fb, 256, 0, stream>>>(bufA, denom, b0, bufZ, N);            // h0 -> bufZ

    // ---------------- GAT layer 1 ----------------
    k_gemm128_wmma<<<(N + 15) / 16, 256, 0, stream>>>(bufZ, W1, bufA, N);    // z1 -> bufA
    k_dots<<<N, HID, 0, stream>>>(bufA, as1, ad1, es, ed, N);
    k_init_node<<<(N + 255) / 256, 256, 0, stream>>>(amax, denom, N);
    k_zero<<<2048, 256, 0, stream>>>(bufZ, nh);                              // accum1 (h0 dead)
    k_edge_max<<<eb, 256, 0, stream>>>(ei, E, N, es, ed, amax);
    k_edge_accum<<<wb, 256, 0, stream>>>(ei, E, N, es, ed, amax, bufA, denom, bufZ);
    k_finalize<<<fb, 256, 0, stream>>>(bufZ, denom, b1, bufA, N);            // h1 -> bufA

    // ---------------- actor MLP + softmax ----------------
    k_mlp_wmma<<<(BV + 15) / 16, 256, 0, stream>>>(bufA, cidx, mW0, mb0, mW1, mb1, scores, BV);
    k_softmax16<<<((BV / 16) + 255) / 256, 256, 0, stream>>>(scores, (float*)d_out, BV / 16);
}